// NestedVectorQuantizer_61125974556917
// MI455X (gfx1250) — compile-verified
//
#include <hip/hip_runtime.h>

#define BB   8192
#define HH   1024
#define DCC  128
#define NC   64      // coarse codes (K0)
#define NF   256     // fine codes per group (K1)
#define KF   1152    // H + DC
#define MAXT 512     // worst-case 16-row tiles per group (8192/16)

typedef float v2f __attribute__((ext_vector_type(2)));
typedef float v8f __attribute__((ext_vector_type(8)));

__device__ __forceinline__ v8f wmma4(v2f a, v2f b, v8f c) {
  // V_WMMA_F32_16X16X4_F32 : D = A(16x4) * B(4x16) + C(16x16), all fp32
  return __builtin_amdgcn_wmma_f32_16x16x4_f32(false, a, false, b, (short)0, c,
                                               false, false);
}

__device__ __forceinline__ float wave_sum(float v) {
#pragma unroll
  for (int off = 16; off; off >>= 1) v += __shfl_xor(v, off, 32);
  return v;
}

// ---------------------------------------------------------------------------
// GEMM: Y[B][128] = concat(A1[B][k1len], A2[B][k2len]) @ W[128][Ktot]^T + bias
// block = 256 thr (8 waves). Block tile: 16 rows x 128 cols; wave w -> 16 cols.
// ---------------------------------------------------------------------------
__global__ __launch_bounds__(256) void gemm16_kernel(
    const float* __restrict__ A1, int k1len, const float* __restrict__ A2,
    int k2len, const float* __restrict__ W, int ldw,
    const float* __restrict__ bias, float* __restrict__ Y) {
  __shared__ float at[16 * 68];  // 16 rows x 64 k, stride 68 (conflict-free b64)
  const int tid = threadIdx.x;
  const int lane = tid & 31;
  const int wv = tid >> 5;
  const int row0 = blockIdx.x * 16;
  const int col0 = wv * 16;
  const int Ktot = k1len + k2len;
  const int n = lane & 15;
  const int koff = (lane >> 4) * 2;  // lanes 0-15: K0,K1 ; lanes 16-31: K2,K3
  v8f acc = {0.f, 0.f, 0.f, 0.f, 0.f, 0.f, 0.f, 0.f};
  const float* wrow = W + (size_t)(col0 + n) * ldw + koff;

  for (int c0 = 0; c0 < Ktot; c0 += 64) {
    __syncthreads();
    {  // cooperative coalesced stage of A chunk (16 rows x 64 k)
      int r = tid >> 4, c4 = tid & 15;
      int kk = c0 + c4 * 4;
      const float* src = (kk < k1len)
                             ? (A1 + (size_t)(row0 + r) * k1len + kk)
                             : (A2 + (size_t)(row0 + r) * k2len + (kk - k1len));
      *(float4*)&at[r * 68 + c4 * 4] = *(const float4*)src;
    }
    __syncthreads();
    const float* ap = &at[(lane & 15) * 68 + koff];
    const float* wp = wrow + c0;
#pragma unroll
    for (int kk = 0; kk < 64; kk += 4) {
      v2f a = *(const v2f*)(ap + kk);   // ds_load_b64, conflict-free
      v2f b = *(const v2f*)(wp + kk);   // B[k][n] = W[col0+n][k]
      acc = wmma4(a, b, acc);
    }
  }
  const int colw = col0 + n;
  const float bv = bias[colw];
  const int rbase = row0 + (lane >> 4) * 8;  // C: vgpr r -> M=r / M=r+8
#pragma unroll
  for (int r = 0; r < 8; ++r)
    Y[(size_t)(rbase + r) * DCC + colw] = acc[r] + bv;
}

// ---------------------------------------------------------------------------
// Coarse: fused LayerNorm + 64-code argmin + zcq_st output + loss partial.
// block = 256 thr = 8 waves = 8 rows. Codebook staged in LDS (pad 133 ->
// conflict-free lanes-over-codes access).
// ---------------------------------------------------------------------------
__global__ __launch_bounds__(256) void coarse_kernel(
    const float* __restrict__ Y, const float* __restrict__ gc,
    const float* __restrict__ betac, const float* __restrict__ cb,
    float* __restrict__ zcq_out, int* __restrict__ ci,
    int* __restrict__ counts, float* __restrict__ partial) {
  __shared__ float cbs[NC * 133];
  __shared__ float zbuf[8 * 128];
  __shared__ float wpart[8];
  const int tid = threadIdx.x, lane = tid & 31, wv = tid >> 5;
  for (int i = tid; i < NC * 128; i += 256)
    cbs[(i >> 7) * 133 + (i & 127)] = cb[i];

  const int row = blockIdx.x * 8 + wv;
  float4 y = *(const float4*)(Y + (size_t)row * 128 + lane * 4);
  float mu = wave_sum(y.x + y.y + y.z + y.w) * (1.f / 128.f);
  float d0 = y.x - mu, d1 = y.y - mu, d2 = y.z - mu, d3 = y.w - mu;
  float var = wave_sum(d0 * d0 + d1 * d1 + d2 * d2 + d3 * d3) * (1.f / 128.f);
  float rstd = rsqrtf(var + 1e-5f);
  float4 g = *(const float4*)(gc + lane * 4);
  float4 be = *(const float4*)(betac + lane * 4);
  float z0 = d0 * rstd * g.x + be.x;
  float z1 = d1 * rstd * g.y + be.y;
  float z2 = d2 * rstd * g.z + be.z;
  float z3 = d3 * rstd * g.w + be.w;
  float* zp = &zbuf[wv * 128 + lane * 4];
  zp[0] = z0; zp[1] = z1; zp[2] = z2; zp[3] = z3;
  __syncthreads();

  float best = 3.4e38f;
  int bidx = 0;
  const float* zr = &zbuf[wv * 128];
#pragma unroll
  for (int c2 = 0; c2 < 2; ++c2) {
    int code = lane + c2 * 32;
    const float* cp = &cbs[code * 133];
    float dot = 0.f, nn = 0.f;
    for (int d = 0; d < 128; ++d) {
      float w = cp[d];
      dot += zr[d] * w;
      nn += w * w;
    }
    float sc = nn - 2.f * dot;  // argmin(||c||^2 - 2 z.c) == argmin d2
    if (sc < best) { best = sc; bidx = code; }
  }
#pragma unroll
  for (int off = 16; off; off >>= 1) {
    float ob = __shfl_xor(best, off, 32);
    int oi = __shfl_xor(bidx, off, 32);
    if (ob < best || (ob == best && oi < bidx)) { best = ob; bidx = oi; }
  }
  if (lane == 0) {
    ci[row] = bidx;
    atomicAdd(&counts[bidx], 1);
  }
  const float* cw = &cbs[bidx * 133 + lane * 4];
  float q0 = cw[0], q1 = cw[1], q2 = cw[2], q3 = cw[3];
  float4 outv = {q0, q1, q2, q3};  // zcq_st == coarse_cb[ci] numerically
  *(float4*)(zcq_out + (size_t)row * 128 + lane * 4) = outv;
  float e0 = q0 - z0, e1 = q1 - z1, e2 = q2 - z2, e3 = q3 - z3;
  float ls = wave_sum(e0 * e0 + e1 * e1 + e2 * e2 + e3 * e3);
  if (lane == 0) wpart[wv] = ls;
  __syncthreads();
  if (tid == 0) {
    float s = 0.f;
#pragma unroll
    for (int i = 0; i < 8; ++i) s += wpart[i];
    partial[blockIdx.x] = s;
  }
}

// LayerNorm only (fine branch): zf = LN(y)*gf + betaf, row-major out.
__global__ __launch_bounds__(256) void ln_kernel(const float* __restrict__ Y,
                                                 const float* __restrict__ gg,
                                                 const float* __restrict__ bb,
                                                 float* __restrict__ Z) {
  const int lane = threadIdx.x & 31, wv = threadIdx.x >> 5;
  const int row = blockIdx.x * 8 + wv;
  float4 y = *(const float4*)(Y + (size_t)row * 128 + lane * 4);
  float mu = wave_sum(y.x + y.y + y.z + y.w) * (1.f / 128.f);
  float d0 = y.x - mu, d1 = y.y - mu, d2 = y.z - mu, d3 = y.w - mu;
  float var = wave_sum(d0 * d0 + d1 * d1 + d2 * d2 + d3 * d3) * (1.f / 128.f);
  float rstd = rsqrtf(var + 1e-5f);
  float4 g = *(const float4*)(gg + lane * 4);
  float4 be = *(const float4*)(bb + lane * 4);
  float4 z = {d0 * rstd * g.x + be.x, d1 * rstd * g.y + be.y,
              d2 * rstd * g.z + be.z, d3 * rstd * g.w + be.w};
  *(float4*)(Z + (size_t)row * 128 + lane * 4) = z;
}

// Per-code squared norms of fine codebooks: norms[64*256]
__global__ void norms_kernel(const float* __restrict__ fcb,
                             float* __restrict__ norms) {
  int id = blockIdx.x * 256 + threadIdx.x;  // 0..16383
  const float* p = fcb + (size_t)id * 128;
  float s = 0.f;
#pragma unroll
  for (int d = 0; d < 128; d += 4) {
    float4 v = *(const float4*)(p + d);
    s += v.x * v.x + v.y * v.y + v.z * v.z + v.w * v.w;
  }
  norms[id] = s;
}

__global__ void init_kernel(int* __restrict__ counts) {
  counts[threadIdx.x] = 0;
}

__global__ void prefix_kernel(const int* __restrict__ counts,
                              int* __restrict__ offs, int* __restrict__ cursor) {
  if (threadIdx.x == 0) {
    int acc = 0;
    for (int g2 = 0; g2 < NC; ++g2) {
      offs[g2] = acc;
      cursor[g2] = acc;
      acc += counts[g2];
    }
    offs[NC] = acc;
  }
}

__global__ void scatter_kernel(const int* __restrict__ ci,
                               int* __restrict__ cursor,
                               int* __restrict__ rowidx) {
  int b = blockIdx.x * 256 + threadIdx.x;
  int g2 = ci[b];
  int pos = atomicAdd(&cursor[g2], 1);
  rowidx[pos] = b;
}

// ---------------------------------------------------------------------------
// Fine: grouped WMMA distances. Block = (group g, 16-row tile t):
// scores[16x256] = -2 * Zf_tile(16x128) @ cb_g(256x128)^T + ||c||^2, argmin,
// writes zfq_st rows and loss partial. Inactive blocks write 0 partial.
// ---------------------------------------------------------------------------
__global__ __launch_bounds__(256) void fine_kernel(
    const float* __restrict__ zf, const float* __restrict__ fcb,
    const float* __restrict__ norms, const int* __restrict__ counts,
    const int* __restrict__ offs, const int* __restrict__ rowidx,
    float* __restrict__ zfq_out, float* __restrict__ partial) {
  const int g2 = blockIdx.x, t = blockIdx.y;
  const int tid = threadIdx.x, lane = tid & 31, wv = tid >> 5;
  const int pidx = t * NC + g2;
  const int ng = counts[g2];
  if (t * 16 >= ng) {
    if (tid == 0) partial[pidx] = 0.f;
    return;
  }
  __shared__ float zt[16 * 132];   // gathered z_f rows (conflict-free b64)
  __shared__ float sc[16 * 260];   // scores [row][code], pad 260
  __shared__ float nrm[NF];
  __shared__ int gr[16];
  __shared__ int gvalid[16];
  __shared__ float wpart[8];
  const int start = offs[g2];
  if (tid < 16) {
    int i = t * 16 + tid;
    int ii = (i < ng) ? i : (ng - 1);  // pad by duplicating last valid row
    gr[tid] = rowidx[start + ii];
    gvalid[tid] = (i < ng);
  }
  nrm[tid] = norms[g2 * NF + tid];
  __syncthreads();
  for (int i = tid; i < 512; i += 256) {  // stage 16x128 gathered A tile
    int r = i >> 5, c4 = i & 31;
    *(float4*)&zt[r * 132 + c4 * 4] =
        *(const float4*)(zf + (size_t)gr[r] * 128 + c4 * 4);
  }
  __syncthreads();

  const int n = lane & 15;
  const int koff = (lane >> 4) * 2;
  v8f acc0 = {0.f, 0.f, 0.f, 0.f, 0.f, 0.f, 0.f, 0.f};
  v8f acc1 = acc0;
  const float* b0 = fcb + ((size_t)g2 * NF + wv * 32 + n) * 128 + koff;
  const float* b1 = b0 + 16 * 128;
  const float* ap = &zt[(lane & 15) * 132 + koff];
#pragma unroll
  for (int k0 = 0; k0 < 128; k0 += 4) {
    v2f a = *(const v2f*)(ap + k0);
    v2f bv0 = *(const v2f*)(b0 + k0);
    v2f bv1 = *(const v2f*)(b1 + k0);
    acc0 = wmma4(a, bv0, acc0);
    acc1 = wmma4(a, bv1, acc1);
  }
#pragma unroll
  for (int r = 0; r < 8; ++r) {
    int rw = r + (lane >> 4) * 8;
    int c0 = wv * 32 + n;
    sc[rw * 260 + c0] = nrm[c0] - 2.f * acc0[r];
    sc[rw * 260 + c0 + 16] = nrm[c0 + 16] - 2.f * acc1[r];
  }
  __syncthreads();

  float lsum = 0.f;
#pragma unroll
  for (int rr = 0; rr < 2; ++rr) {  // wave wv handles rows 2wv, 2wv+1
    int rw = wv * 2 + rr;
    float best = 3.4e38f;
    int bidx = 0;
    const float* sp = &sc[rw * 260];
#pragma unroll
    for (int j = 0; j < 8; ++j) {
      int code = lane + j * 32;
      float v = sp[code];
      if (v < best) { best = v; bidx = code; }
    }
#pragma unroll
    for (int off = 16; off; off >>= 1) {
      float ob = __shfl_xor(best, off, 32);
      int oi = __shfl_xor(bidx, off, 32);
      if (ob < best || (ob == best && oi < bidx)) { best = ob; bidx = oi; }
    }
    if (gvalid[rw]) {
      int grow = gr[rw];
      const float* q = fcb + ((size_t)g2 * NF + bidx) * 128 + lane * 4;
      float4 qv = *(const float4*)q;  // zfq_st == fine_cbs[ci,fi]
      *(float4*)(zfq_out + (size_t)grow * 128 + lane * 4) = qv;
      float e0 = qv.x - zt[rw * 132 + lane * 4 + 0];
      float e1 = qv.y - zt[rw * 132 + lane * 4 + 1];
      float e2 = qv.z - zt[rw * 132 + lane * 4 + 2];
      float e3 = qv.w - zt[rw * 132 + lane * 4 + 3];
      lsum += e0 * e0 + e1 * e1 + e2 * e2 + e3 * e3;
    }
  }
  float ls = wave_sum(lsum);
  if (lane == 0) wpart[wv] = ls;
  __syncthreads();
  if (tid == 0) {
    float s = 0.f;
#pragma unroll
    for (int i = 0; i < 8; ++i) s += wpart[i];
    partial[pidx] = s;
  }
}

// Final loss: 1.25 * (sum of partials) / (B*128), fixed-order reduction.
__global__ void loss_kernel(const float* __restrict__ pc, int npc,
                            const float* __restrict__ pf, int npf,
                            float* __restrict__ out) {
  __shared__ float red[256];
  int tid = threadIdx.x;
  float s = 0.f;
  for (int i = tid; i < npc; i += 256) s += pc[i];
  for (int i = tid; i < npf; i += 256) s += pf[i];
  red[tid] = s;
  __syncthreads();
  for (int st = 128; st; st >>= 1) {
    if (tid < st) red[tid] += red[tid + st];
    __syncthreads();
  }
  if (tid == 0) out[0] = 1.25f * red[0] / (8192.0f * 128.0f);
}

extern "C" void kernel_launch(void* const* d_in, const int* in_sizes, int n_in,
                              void* d_out, int out_size, void* d_ws,
                              size_t ws_size, hipStream_t stream) {
  (void)in_sizes; (void)n_in; (void)out_size; (void)ws_size;
  const float* x     = (const float*)d_in[0];
  const float* Wc    = (const float*)d_in[1];
  const float* bc    = (const float*)d_in[2];
  const float* gc    = (const float*)d_in[3];
  const float* betac = (const float*)d_in[4];
  const float* ccb   = (const float*)d_in[5];
  const float* Wf    = (const float*)d_in[6];
  const float* bf    = (const float*)d_in[7];
  const float* gf    = (const float*)d_in[8];
  const float* betaf = (const float*)d_in[9];
  const float* fcb   = (const float*)d_in[10];

  float* out  = (float*)d_out;
  float* zcq  = out;                      // [B][128] zcq_st
  float* zfq  = out + (size_t)BB * DCC;   // [B][128] zfq_st
  float* loss = out + (size_t)2 * BB * DCC;

  // workspace layout (~8.7 MB)
  float* y     = (float*)d_ws;            // [B][128] pre-LN (reused)
  float* zf    = y + (size_t)BB * DCC;    // [B][128] z_f
  float* norms = zf + (size_t)BB * DCC;   // [64*256]
  float* pco   = norms + NC * NF;         // [1024] coarse loss partials
  float* pfi   = pco + (BB / 8);          // [64*512] fine loss partials
  int* ci      = (int*)(pfi + NC * MAXT); // [B]
  int* counts  = ci + BB;                 // [64]
  int* offs    = counts + NC;             // [65]
  int* cursor  = offs + NC + 1;           // [64]
  int* rowidx  = cursor + NC;             // [B]

  init_kernel<<<1, NC, 0, stream>>>(counts);
  gemm16_kernel<<<BB / 16, 256, 0, stream>>>(x, HH, nullptr, 0, Wc, HH, bc, y);
  coarse_kernel<<<BB / 8, 256, 0, stream>>>(y, gc, betac, ccb, zcq, ci, counts,
                                            pco);
  gemm16_kernel<<<BB / 16, 256, 0, stream>>>(x, HH, zcq, DCC, Wf, KF, bf, y);
  ln_kernel<<<BB / 8, 256, 0, stream>>>(y, gf, betaf, zf);
  norms_kernel<<<NC, 256, 0, stream>>>(fcb, norms);
  prefix_kernel<<<1, 1, 0, stream>>>(counts, offs, cursor);
  scatter_kernel<<<BB / 256, 256, 0, stream>>>(ci, cursor, rowidx);
  fine_kernel<<<dim3(NC, MAXT), 256, 0, stream>>>(zf, fcb, norms, counts, offs,
                                                  rowidx, zfq, pfi);
  loss_kernel<<<1, 256, 0, stream>>>(pco, BB / 8, pfi, NC * MAXT, loss);
}